// VirtualEmbeddingV5_22874995818884
// MI455X (gfx1250) — compile-verified
//
#include <hip/hip_runtime.h>
#include <hip/hip_bf16.h>
#include <math.h>

// ---- problem dims (fixed by reference) ----
#define D_EMB   768
#define A_PAD   16
#define LANG_W  32
#define EMB_W   (D_EMB + A_PAD + LANG_W + D_EMB)   // 1584
#define SYMPAD_W (D_EMB + A_PAD)                   // 784
#define S_SYN   8192
#define V_VOCA  32000
#define KMAX    4
#define BL_TOK  4096                               // B*L = 2*2048

typedef float v2f __attribute__((ext_vector_type(2)));
typedef float v8f __attribute__((ext_vector_type(8)));
typedef unsigned int u32x4 __attribute__((ext_vector_type(4)));
typedef int i32x4 __attribute__((ext_vector_type(4)));
typedef int i32x8 __attribute__((ext_vector_type(8)));

__device__ __forceinline__ float sigmoid_f(float z) {
  return 1.0f / (1.0f + __expf(-z));
}

// Low 32 bits of a flat shared-aperture address == LDS byte offset.
__device__ __forceinline__ unsigned lds_off(const void* p) {
  return (unsigned)(size_t)p;
}

// ---------------------------------------------------------------------------
// TDM 2D tile load: tile_h rows x tile_w f32 elements, global row stride
// row_stride elements, into LDS at lds_byte_off. pad17: insert 1 dword of LDS
// padding after every 16 dwords stored (=> 17-float LDS row stride for a
// 16-float-wide tile). Descriptor packing per CDNA5 ISA 8.3/8.4; groups 2/3
// zero (2D tensor: tile_dim2=0, iterate disabled). Tracked by TENSORcnt.
// ---------------------------------------------------------------------------
__device__ __forceinline__ void tdm_load_2d_f32(
    const float* gptr, unsigned lds_byte_off,
    unsigned tile_w, unsigned tile_h, unsigned row_stride, int pad17) {
  const unsigned long long ga = (unsigned long long)(size_t)gptr;

  u32x4 g0;
  g0.x = 0x1u;                                   // count=1 (valid user D#)
  g0.y = lds_byte_off;                           // lds_addr [63:32]
  g0.z = (unsigned)(ga & 0xFFFFFFFFu);           // global_addr[31:0]
  g0.w = (unsigned)((ga >> 32) & 0x1FFFFFFu)     // global_addr[56:32]
       | (2u << 30);                             // type=2 ("image")

  unsigned d0 = (2u << 16);                      // data_size=2 -> 4 bytes
  if (pad17) d0 |= (1u << 20)                    // pad_enable
             | (3u << 22)                        // pad_interval: 16 dwords
             | (0u << 25);                       // pad_amount: 1 dword
  i32x8 g1;
  g1[0] = (int)d0;                               // wg_mask=0, flags
  g1[1] = (int)((tile_w & 0xFFFFu) << 16);       // tensor_dim0[15:0]
  g1[2] = (int)(((tile_w >> 16) & 0xFFFFu)       // tensor_dim0[31:16]
       | ((tile_h & 0xFFFFu) << 16));            // tensor_dim1[15:0]
  g1[3] = (int)(((tile_h >> 16) & 0xFFFFu)       // tensor_dim1[31:16]
       | ((tile_w & 0xFFFFu) << 16));            // tile_dim0
  g1[4] = (int)(tile_h & 0xFFFFu);               // tile_dim1 (tile_dim2=0)
  g1[5] = (int)row_stride;                       // tensor_dim0_stride[31:0]
  g1[6] = 0;                                     // stride hi / dim1_stride lo
  g1[7] = 0;                                     // dim1_stride hi (unused, 2D)

  const i32x4 z4 = {0, 0, 0, 0};
  const i32x8 z8 = {0, 0, 0, 0, 0, 0, 0, 0};
  // 6-arg toolchain form: (g0, g1, g2, g3, extra, cpol)
  __builtin_amdgcn_tensor_load_to_lds(g0, g1, z4, z4, z8, 0);
}

// ---------------------------------------------------------------------------
// Stage 1: per-token gather + three_stage activation + pad constants.
// emb row: [0..767]=W1[id], [768..815]=0.1, [816..1583]=syn
// ---------------------------------------------------------------------------
__global__ __launch_bounds__(256) void k_embed_gather(
    const int* __restrict__ ids, const float* __restrict__ W1,
    const float* __restrict__ W2, float* __restrict__ emb) {
  const int t = blockIdx.x;
  const int id = ids[t];
  const float* w1 = W1 + (size_t)id * D_EMB;
  const float* w2 = W2 + (size_t)id * D_EMB;
  float* e = emb + (size_t)t * EMB_W;
  for (int i = threadIdx.x; i < D_EMB; i += 256) {
    e[i] = w1[i];
    const float x = w2[i];  // (W2[id]/SCALE)*SCALE == W2[id]
    const float s = sigmoid_f(3.0f * (x - 4.0f)) - sigmoid_f(3.0f * (-x - 4.0f));
    e[(D_EMB + A_PAD + LANG_W) + i] = s;
  }
  if (threadIdx.x < A_PAD + LANG_W) e[D_EMB + threadIdx.x] = 0.1f;
}

// ---------------------------------------------------------------------------
// fp32 WMMA GEMM with TDM double-buffered LDS staging.
//   C[M,N] = (accum ? C : 0) + A[M,K](lda) * B[K,N](ldb) + bias[N]
// Requires M%64==0, N%128==0, K%16==0.
// Block: 256 threads / 8 waves; block tile 64x128; wave tile 16x64.
// Per stage, wave 0 issues two TDM descriptors (A 64x16 with HW-padded
// 17-float LDS stride; B 16x128 dense), then s_wait_tensorcnt + barrier.
// ---------------------------------------------------------------------------
__global__ __launch_bounds__(256) void k_wmma_gemm_f32(
    const float* __restrict__ A, int lda,
    const float* __restrict__ B, int ldb,
    const float* __restrict__ bias,
    float* __restrict__ C, int ldc,
    int K, int accum) {
  __shared__ float Alds[2][64 * 17];   // 17-float row stride via TDM padding
  __shared__ float Blds[2][16 * 128];

  const int tid  = threadIdx.x;
  const int wave = tid >> 5;
  const int lane = tid & 31;
  const int lm   = lane & 15;
  const int lh   = lane >> 4;
  const int tm   = wave & 3;    // M sub-tile 0..3
  const int tn   = wave >> 2;   // N half 0..1

  const int m0 = blockIdx.y * 64;
  const int n0 = blockIdx.x * 128;

  v8f zero = {};
  v8f acc[4];
  for (int j = 0; j < 4; ++j) acc[j] = zero;

  const int nstages = K >> 4;
  const bool issuer = (wave == 0);   // uniform per wave: EXEC all-1s inside

  auto issue = [&](int s, int p) {
    const float* Ag = &A[(size_t)m0 * lda + (s << 4)];
    const float* Bg = &B[(size_t)(s << 4) * ldb + n0];
    tdm_load_2d_f32(Ag, lds_off(&Alds[p][0]), 16, 64, (unsigned)lda, 1);
    tdm_load_2d_f32(Bg, lds_off(&Blds[p][0]), 128, 16, (unsigned)ldb, 0);
  };

  if (issuer) issue(0, 0);

  for (int s = 0; s < nstages; ++s) {
    const int p = s & 1;
    if (issuer) {
      if (s + 1 < nstages) {
        issue(s + 1, p ^ 1);                    // prefetch next stage
        __builtin_amdgcn_s_wait_tensorcnt(2);   // pair for buf p complete
      } else {
        __builtin_amdgcn_s_wait_tensorcnt(0);
      }
    }
    __syncthreads();   // buf p visible to all waves

#pragma unroll
    for (int kk = 0; kk < 16; kk += 4) {
      const float* ap = &Alds[p][(tm * 16 + lm) * 17 + kk + lh * 2];
      v2f a; a.x = ap[0]; a.y = ap[1];
#pragma unroll
      for (int j = 0; j < 4; ++j) {
        const float* bp = &Blds[p][(kk + lh * 2) * 128 + (tn * 64 + j * 16 + lm)];
        v2f b; b.x = bp[0]; b.y = bp[128];
        acc[j] = __builtin_amdgcn_wmma_f32_16x16x4_f32(
            false, a, false, b, (short)0, acc[j], false, false);
      }
    }
    __syncthreads();   // buf p consumed; safe to overwrite next iteration
  }

  // epilogue: bias add, optional accumulate into existing C
#pragma unroll
  for (int j = 0; j < 4; ++j) {
    const int n = n0 + tn * 64 + j * 16 + lm;
    const float bv = bias ? bias[n] : 0.0f;
#pragma unroll
    for (int r = 0; r < 8; ++r) {
      const int m = m0 + tm * 16 + lh * 8 + r;
      const size_t off = (size_t)m * ldc + n;
      float v = acc[j][r] + bv;
      if (accum) v += C[off];
      C[off] = v;
    }
  }
}

// ---------------------------------------------------------------------------
// expected_synonym_id_sum: per-row zero + K=4 serial scatter adds (duplicate
// synonym ids accumulate like jnp .at[].add)
// ---------------------------------------------------------------------------
__global__ __launch_bounds__(256) void k_expected_hist(
    const int* __restrict__ ids, const int* __restrict__ syn_table,
    const float* __restrict__ syn_mask, float* __restrict__ expected) {
  const int t = blockIdx.x;
  float* row = expected + (size_t)t * S_SYN;
  for (int i = threadIdx.x; i < S_SYN; i += 256) row[i] = 0.0f;
  __syncthreads();
  if (threadIdx.x == 0) {
    const int id = ids[t];
#pragma unroll
    for (int k = 0; k < KMAX; ++k) {
      row[syn_table[id * KMAX + k]] += syn_mask[id * KMAX + k];
    }
  }
}

// ---------------------------------------------------------------------------
extern "C" void kernel_launch(void* const* d_in, const int* in_sizes, int n_in,
                              void* d_out, int out_size, void* d_ws, size_t ws_size,
                              hipStream_t stream) {
  const int*   ids       = (const int*)d_in[0];
  const float* W1        = (const float*)d_in[1];
  const float* W2        = (const float*)d_in[2];
  const float* W_se      = (const float*)d_in[3];
  const float* b_se      = (const float*)d_in[4];
  const float* W_ss      = (const float*)d_in[5];
  const float* b_ss      = (const float*)d_in[6];
  const float* W_rev     = (const float*)d_in[7];
  const int*   syn_table = (const int*)d_in[8];
  const float* syn_mask  = (const float*)d_in[9];

  float* out      = (float*)d_out;
  float* emb      = out;                               // [4096,1584]
  float* logits   = emb    + (size_t)BL_TOK * EMB_W;   // [4096,32000]
  float* syn_sum  = logits + (size_t)BL_TOK * V_VOCA;  // [4096,8192]
  float* expected = syn_sum + (size_t)BL_TOK * S_SYN;  // [4096,8192]

  // 1) gather + activation + pads -> emb
  k_embed_gather<<<BL_TOK, 256, 0, stream>>>(ids, W1, W2, emb);

  // 2) symbol += syn @ W_se + b_se   (A = emb syn cols, C = emb symbol cols)
  k_wmma_gemm_f32<<<dim3(D_EMB / 128, BL_TOK / 64), 256, 0, stream>>>(
      emb + (D_EMB + A_PAD + LANG_W), EMB_W, W_se, D_EMB, b_se,
      emb, EMB_W, /*K=*/D_EMB, /*accum=*/1);

  // 3) logits = symbol_padded @ W_rev   (A = emb cols 0..783, K=784)
  k_wmma_gemm_f32<<<dim3(V_VOCA / 128, BL_TOK / 64), 256, 0, stream>>>(
      emb, EMB_W, W_rev, V_VOCA, nullptr,
      logits, V_VOCA, /*K=*/SYMPAD_W, /*accum=*/0);

  // 4) syn_sum = syn @ W_ss + b_ss
  k_wmma_gemm_f32<<<dim3(S_SYN / 128, BL_TOK / 64), 256, 0, stream>>>(
      emb + (D_EMB + A_PAD + LANG_W), EMB_W, W_ss, S_SYN, b_ss,
      syn_sum, S_SYN, /*K=*/D_EMB, /*accum=*/0);

  // 5) expected histogram
  k_expected_hist<<<BL_TOK, 256, 0, stream>>>(ids, syn_table, syn_mask, expected);
}